// ViewLearner_79139067396168
// MI455X (gfx1250) — compile-verified
//
#include <hip/hip_runtime.h>

#define EMB_D 128
#define MLP_D 64

typedef float v2f __attribute__((ext_vector_type(2)));
typedef float v8f __attribute__((ext_vector_type(8)));

// ---------------------------------------------------------------------------
// Kernel 1: C[N,128] = node_emb[N,128] @ Wcat[128,128]
//   Wcat[:, 0:64]  = W1[0:128, :]    (src projection)
//   Wcat[:, 64:128] = W1[128:256, :] (dst projection)
// One wave per 16x16 output tile using V_WMMA_F32_16X16X4_F32.
// ---------------------------------------------------------------------------
__global__ __launch_bounds__(256) void proj_gemm_wmma(
    const float* __restrict__ emb,   // [N, 128]
    const float* __restrict__ W1,    // [256, 64] row-major
    float* __restrict__ Cout,        // [N, 128]
    int N)
{
    const int lane = threadIdx.x & 31;
    const int wave = threadIdx.x >> 5;          // 0..7 -> column tile
    const int m0   = blockIdx.x * 16;           // row-tile base
    const int n0   = wave * 16;                 // col-tile base
    const int half = lane >> 4;                 // 0: lanes 0-15, 1: lanes 16-31
    const int l15  = lane & 15;

    // A-matrix row this lane feeds (clamped for generic N; N=100000 is 16-aligned)
    int arow = m0 + l15;
    if (arow >= N) arow = N - 1;
    const float* aRow = emb + (size_t)arow * EMB_D;

    // B column this lane feeds: Wcat[k][n0+l15]
    const int ncol = n0 + l15;
    const float* wcol = (ncol < MLP_D) ? (W1 + ncol)
                                       : (W1 + (size_t)EMB_D * MLP_D + (ncol - MLP_D));

    v8f acc = {};
#pragma unroll
    for (int kk = 0; kk < EMB_D; kk += 4) {
        // ISA 16x4 fp32 A layout: lanes 0-15 hold K={kk,kk+1}, lanes 16-31 hold K={kk+2,kk+3}
        const int ka = kk + 2 * half;
        v2f a, b;
        a.x = aRow[ka + 0];
        a.y = aRow[ka + 1];
        // ISA 4x16 fp32 B layout mirrors A: lanes 0-15 rows K={kk,kk+1}, lanes 16-31 K={kk+2,kk+3}
        b.x = wcol[(size_t)(ka + 0) * MLP_D];
        b.y = wcol[(size_t)(ka + 1) * MLP_D];
        acc = __builtin_amdgcn_wmma_f32_16x16x4_f32(
            /*neg_a=*/false, a, /*neg_b=*/false, b,
            /*c_mod=*/(short)0, acc, /*reuse_a=*/false, /*reuse_b=*/false);
    }

    // 16x16 f32 D layout: lanes 0-15 -> rows m0+0..7 (vgpr r = row r),
    //                     lanes 16-31 -> rows m0+8..15; column = n0 + l15.
    const int rbase = m0 + half * 8;
#pragma unroll
    for (int r = 0; r < 8; ++r) {
        const int row = rbase + r;
        if (row < N) Cout[(size_t)row * EMB_D + n0 + l15] = acc[r];
    }
}

// ---------------------------------------------------------------------------
// Kernel 2: per-edge MLP from precomputed projections.
// One wave per edge; lane j handles hidden units {2j, 2j+1}.
//   logit = W2 . relu(C[src][0:64] + C[dst][64:128] + b1) + b2
// ---------------------------------------------------------------------------
__global__ __launch_bounds__(256) void edge_mlp(
    const float* __restrict__ C,      // [N, 128]
    const int*  __restrict__ src,
    const int*  __restrict__ dst,
    const float* __restrict__ b1,     // [64]
    const float* __restrict__ W2,     // [64]
    const float* __restrict__ b2,     // [1]
    float* __restrict__ out,          // [E]
    int E)
{
    const int lane = threadIdx.x & 31;
    const int e = blockIdx.x * (blockDim.x >> 5) + (threadIdx.x >> 5);
    if (e >= E) return;

    const int s = src[e];
    const int d = dst[e];

    // 256B fully-coalesced gathers (L2-resident: C is 51.2MB < 192MB L2)
    const float2 av   = ((const float2*)(C + (size_t)s * EMB_D))[lane];
    const float2 bv   = ((const float2*)(C + (size_t)d * EMB_D + MLP_D))[lane];
    const float2 bias = ((const float2*)b1)[lane];
    const float2 w    = ((const float2*)W2)[lane];

    float h0 = av.x + bv.x + bias.x; h0 = h0 > 0.0f ? h0 : 0.0f;
    float h1 = av.y + bv.y + bias.y; h1 = h1 > 0.0f ? h1 : 0.0f;
    float v = h0 * w.x + h1 * w.y;

    // wave32 reduction
#pragma unroll
    for (int off = 16; off > 0; off >>= 1)
        v += __shfl_xor(v, off, 32);

    if (lane == 0) out[e] = v + b2[0];
}

// ---------------------------------------------------------------------------
// Fallback (only if workspace is too small): direct per-edge MLP.
// ---------------------------------------------------------------------------
__global__ __launch_bounds__(256) void edge_mlp_direct(
    const float* __restrict__ emb,
    const int*  __restrict__ src,
    const int*  __restrict__ dst,
    const float* __restrict__ W1,
    const float* __restrict__ b1,
    const float* __restrict__ W2,
    const float* __restrict__ b2,
    float* __restrict__ out,
    int E)
{
    const int e = blockIdx.x * blockDim.x + threadIdx.x;
    if (e >= E) return;
    const float* es = emb + (size_t)src[e] * EMB_D;
    const float* ed = emb + (size_t)dst[e] * EMB_D;
    float acc = b2[0];
    for (int j = 0; j < MLP_D; ++j) {
        float h = b1[j];
        for (int k = 0; k < EMB_D; ++k) h += es[k] * W1[(size_t)k * MLP_D + j];
        for (int k = 0; k < EMB_D; ++k) h += ed[k] * W1[(size_t)(EMB_D + k) * MLP_D + j];
        if (h > 0.0f) acc += h * W2[j];
    }
    out[e] = acc;
}

extern "C" void kernel_launch(void* const* d_in, const int* in_sizes, int n_in,
                              void* d_out, int out_size, void* d_ws, size_t ws_size,
                              hipStream_t stream)
{
    const float* node_emb = (const float*)d_in[0];   // [N,128]
    const int*   edge_idx = (const int*)  d_in[1];   // [2,E] flat
    const float* W1       = (const float*)d_in[2];   // [256,64]
    const float* b1       = (const float*)d_in[3];   // [64]
    const float* W2       = (const float*)d_in[4];   // [64,1]
    const float* b2       = (const float*)d_in[5];   // [1]
    float* out = (float*)d_out;

    const int N = in_sizes[0] / EMB_D;   // 100000
    const int E = in_sizes[1] / 2;       // 500000
    const int* src = edge_idx;
    const int* dst = edge_idx + E;

    const size_t need = (size_t)N * EMB_D * sizeof(float);  // 51.2 MB
    if (ws_size >= need) {
        float* C = (float*)d_ws;
        const int rowTiles = (N + 15) / 16;   // 6250
        proj_gemm_wmma<<<rowTiles, 256, 0, stream>>>(node_emb, W1, C, N);

        const int wavesPerBlock = 8;          // 256 threads
        const int blocks = (E + wavesPerBlock - 1) / wavesPerBlock;
        edge_mlp<<<blocks, 256, 0, stream>>>(C, src, dst, b1, W2, b2, out, E);
    } else {
        edge_mlp_direct<<<(E + 255) / 256, 256, 0, stream>>>(
            node_emb, src, dst, W1, b1, W2, b2, out, E);
    }
}